// Roi_61564061221098
// MI455X (gfx1250) — compile-verified
//
#include <hip/hip_runtime.h>
#include <hip/hip_bf16.h>
#include <cstdint>

// Problem constants (from reference setup_inputs)
#define RR   2000
#define NN   4
#define CC   512
#define HH   50
#define WW   68
#define PBIN 7
#define CPB  8          // channels per block == waves per block
#define SCALE 0.0625f

__global__ __launch_bounds__(256) void roi_maxpool_async_kernel(
    const float* __restrict__ x,        // (N,C,H,W)
    const float* __restrict__ rois,     // (R,4) = [y0,x0,y1,x1]
    const int*   __restrict__ roi_idx,  // (R,)
    float*       __restrict__ out)      // (R, C*49)
{
    // per-wave double-buffered staging row: up to ceil(68/32)*32 = 96 slots
    __shared__ float rowbuf[CPB][2][96];

    const int tid  = threadIdx.x;
    const int wv   = tid >> 5;          // wave id 0..7
    const int lane = tid & 31;
    const int r    = blockIdx.y;                 // roi id
    const int c    = blockIdx.x * CPB + wv;      // channel id
    const int n    = roi_idx[r];

    // Reference permutes [y0,x0,y1,x1] -> boxes [x0,y0,x1,y1]
    const float ry0f = rois[r * 4 + 0];
    const float rx0f = rois[r * 4 + 1];
    const float ry1f = rois[r * 4 + 2];
    const float rx1f = rois[r * 4 + 3];
    const float rx0 = rintf(rx0f * SCALE);   // round-half-even == jnp.round
    const float ry0 = rintf(ry0f * SCALE);
    const float rx1 = rintf(rx1f * SCALE);
    const float ry1 = rintf(ry1f * SCALE);
    const float roiW = fmaxf(rx1 - rx0 + 1.0f, 1.0f);
    const float roiH = fmaxf(ry1 - ry0 + 1.0f, 1.0f);
    const float bh = roiH / 7.0f;            // keep exact fp32 divide (matches jnp)
    const float bw = roiW / 7.0f;

    // Union region bounds (bin boundaries are monotone in p)
    const int h0 = (int)fminf(fmaxf(ry0, 0.0f), (float)HH);
    const int h1 = (int)fminf(fmaxf(ceilf(7.0f * bh) + ry0, 0.0f), (float)HH);
    const int w0 = (int)fminf(fmaxf(rx0, 0.0f), (float)WW);
    const int w1 = (int)fminf(fmaxf(ceilf(7.0f * bw) + rx0, 0.0f), (float)WW);
    const int regW = w1 - w0;
    const int regH = h1 - h0;

    // Lane handles bins b0 = lane and b1 = lane+32 (49 bins total)
    const int b0 = lane;
    const int b1 = lane + 32;
    const bool has1 = (b1 < 49);

    int hs0, he0, ws0, we0;
    {
        const float ph = (float)(b0 / 7), pw = (float)(b0 % 7);
        hs0 = (int)fminf(fmaxf(floorf(ph * bh) + ry0, 0.0f), (float)HH);
        he0 = (int)fminf(fmaxf(ceilf((ph + 1.0f) * bh) + ry0, 0.0f), (float)HH);
        ws0 = (int)fminf(fmaxf(floorf(pw * bw) + rx0, 0.0f), (float)WW);
        we0 = (int)fminf(fmaxf(ceilf((pw + 1.0f) * bw) + rx0, 0.0f), (float)WW);
    }
    int hs1 = 0, he1 = 0, ws1 = 0, we1 = 0;
    if (has1) {
        const float ph = (float)(b1 / 7), pw = (float)(b1 % 7);
        hs1 = (int)fminf(fmaxf(floorf(ph * bh) + ry0, 0.0f), (float)HH);
        he1 = (int)fminf(fmaxf(ceilf((ph + 1.0f) * bh) + ry0, 0.0f), (float)HH);
        ws1 = (int)fminf(fmaxf(floorf(pw * bw) + rx0, 0.0f), (float)WW);
        we1 = (int)fminf(fmaxf(ceilf((pw + 1.0f) * bw) + rx0, 0.0f), (float)WW);
    }

    float m0 = -INFINITY, m1 = -INFINITY;

    if (regW > 0 && regH > 0) {
        const int nchunks = (regW + 31) >> 5;       // 1..3 async ops per row
        const float* plane = x + (((size_t)n * CC + c) * HH) * (size_t)WW;

        // Issue one region row (coalesced 128B bursts) into LDS buffer `buf`
        // via the CDNA5 async global->LDS engine (tracked on ASYNCcnt).
        auto issue_row = [&](int h, int buf) {
            // DS reads of the buffer being overwritten were issued last
            // iteration; DS and async ops are unordered -> drain DScnt first.
            asm volatile("s_wait_dscnt 0x0" ::: "memory");
            const float* rowp = plane + (size_t)h * WW;
            for (int k = 0; k < nchunks; ++k) {
                const int i  = lane + (k << 5);
                const int ii = (i < regW) ? i : (regW - 1);  // clamp: uniform issue count
                const uint64_t ga = (uint64_t)(uintptr_t)(rowp + (w0 + ii));
                // low 32 bits of a generic LDS pointer == LDS byte address
                const uint32_t la = (uint32_t)(uintptr_t)(&rowbuf[wv][buf][i]);
                asm volatile("global_load_async_to_lds_b32 %0, %1, off"
                             :: "v"(la), "v"(ga)
                             : "memory");
            }
        };

        issue_row(h0, 0);
        for (int h = h0; h < h1; ++h) {
            const int  cur = (h - h0) & 1;
            const bool pre = (h + 1 < h1);
            if (pre) issue_row(h + 1, cur ^ 1);     // prefetch next row
            if (pre) {
                // async loads complete in order: <=nchunks outstanding
                // means only the just-issued next row can still be in flight.
                if (nchunks == 1)      asm volatile("s_wait_asynccnt 0x1" ::: "memory");
                else if (nchunks == 2) asm volatile("s_wait_asynccnt 0x2" ::: "memory");
                else                   asm volatile("s_wait_asynccnt 0x3" ::: "memory");
            } else {
                asm volatile("s_wait_asynccnt 0x0" ::: "memory");
            }
            const float* rb = &rowbuf[wv][cur][0];
            if (h >= hs0 && h < he0) {
                for (int w = ws0; w < we0; ++w) m0 = fmaxf(m0, rb[w - w0]);
            }
            if (has1 && h >= hs1 && h < he1) {
                for (int w = ws1; w < we1; ++w) m1 = fmaxf(m1, rb[w - w0]);
            }
        }
        // make sure nothing is left outstanding before wave ends
        asm volatile("s_wait_asynccnt 0x0" ::: "memory");
    }

    // -inf (empty bin) -> 0, exactly like jnp.where(isfinite(z), z, 0).
    // Output is a 200MB write-once stream: store non-temporal so it does not
    // displace the fully L2-resident 28MB feature map.
    float* orow = out + ((size_t)r * CC + c) * 49;
    __builtin_nontemporal_store(__builtin_isinf(m0) ? 0.0f : m0, &orow[b0]);
    if (has1)
        __builtin_nontemporal_store(__builtin_isinf(m1) ? 0.0f : m1, &orow[b1]);
}

extern "C" void kernel_launch(void* const* d_in, const int* in_sizes, int n_in,
                              void* d_out, int out_size, void* d_ws, size_t ws_size,
                              hipStream_t stream) {
    (void)in_sizes; (void)n_in; (void)d_ws; (void)ws_size; (void)out_size;
    const float* x       = (const float*)d_in[0];   // (4,512,50,68) fp32
    const float* rois    = (const float*)d_in[1];   // (2000,4) fp32
    const int*   roi_idx = (const int*)d_in[2];     // (2000,) int32
    float*       out     = (float*)d_out;           // (2000, 512*49) fp32

    dim3 grid(CC / CPB, RR);   // (64, 2000)
    dim3 block(256);           // 8 wave32s, one channel per wave
    roi_maxpool_async_kernel<<<grid, block, 0, stream>>>(x, rois, roi_idx, out);
}